// FeatureExtractorGAT_63625645523573
// MI455X (gfx1250) — compile-verified
//
#include <hip/hip_runtime.h>
#include <math.h>

typedef float v2f __attribute__((ext_vector_type(2)));
typedef float v8f __attribute__((ext_vector_type(8)));

#define H1 8
#define C1 60
#define F1 480   // H1*C1
#define F2 96
#define NEG_SLOPE 0.2f

__device__ __forceinline__ v8f wmma_f32_k4(v2f a, v2f b, v8f c) {
    return __builtin_amdgcn_wmma_f32_16x16x4_f32(false, a, false, b, (short)0, c, false, false);
}

// ---------------------------------------------------------------------------
// GEMM1: xh1[N,480] = x[N,4] @ W1[4,480] -- one V_WMMA_F32_16X16X4_F32 per tile
// grid: (ceil(N/16), 480/16), block: 32 (one wave)
// ---------------------------------------------------------------------------
__global__ __launch_bounds__(32) void gemm1_wmma(const float* __restrict__ x,
                                                 const float* __restrict__ W1,
                                                 float* __restrict__ xh, int N) {
    const int lane = threadIdx.x;
    const int half = lane >> 4;     // 0: K={0,1}, D rows 0..7 ; 1: K={2,3}, D rows 8..15
    const int l16  = lane & 15;
    const int row0 = blockIdx.x * 16;
    const int col0 = blockIdx.y * 16;

    // A tile 16x4 f32: lane l16 holds row M=l16; half selects K-pair
    const int row = row0 + l16;
    const int ka  = half * 2;
    v2f a = {0.f, 0.f};
    if (row < N) a = *(const v2f*)(x + (size_t)row * 4 + ka);

    // B tile 4x16 f32: lane l16 holds col N=l16; half selects K-pair
    v2f b;
    b.x = W1[(ka)     * F1 + col0 + l16];
    b.y = W1[(ka + 1) * F1 + col0 + l16];

    v8f c = {};
    c = wmma_f32_k4(a, b, c);

    // D layout: VGPR r, lanes 0-15 -> (M=r, N=l16); lanes 16-31 -> (M=8+r, N=l16)
    if (row0 + 16 <= N) {
#pragma unroll
        for (int r = 0; r < 8; ++r)
            xh[(size_t)(row0 + r + half * 8) * F1 + col0 + l16] = c[r];
    } else {
#pragma unroll
        for (int r = 0; r < 8; ++r) {
            const int m = row0 + r + half * 8;
            if (m < N) xh[(size_t)m * F1 + col0 + l16] = c[r];
        }
    }
}

// ---------------------------------------------------------------------------
// GEMM2: xh2[N,96] = h[N,480] @ W2[480,96]
// One wave per 16-row tile computes ALL six 16-col tiles -> A loaded once,
// 6 independent WMMA accumulator chains (ILP), minimal h traffic.
// grid: ceil(N/16), block: 32
// ---------------------------------------------------------------------------
__global__ __launch_bounds__(32) void gemm2_wmma(const float* __restrict__ h,
                                                 const float* __restrict__ W2,
                                                 float* __restrict__ xh2, int N) {
    const int lane = threadIdx.x;
    const int half = lane >> 4;
    const int l16  = lane & 15;
    const int row0 = blockIdx.x * 16;
    const int row  = row0 + l16;
    const bool rv  = (row < N);
    const float* hrow = h + (size_t)row * F1;

    v8f acc[6];
#pragma unroll
    for (int t = 0; t < 6; ++t) acc[t] = (v8f){};

    for (int k0 = 0; k0 < F1; k0 += 4) {
        const int ka = k0 + half * 2;
        v2f a = {0.f, 0.f};
        if (rv) a = *(const v2f*)(hrow + ka);        // global_load_b64
        const float* w0 = W2 + (size_t)ka * F2 + l16;
        const float* w1 = w0 + F2;
#pragma unroll
        for (int t = 0; t < 6; ++t) {
            v2f b;
            b.x = w0[t * 16];
            b.y = w1[t * 16];
            acc[t] = wmma_f32_k4(a, b, acc[t]);      // 6 independent chains
        }
    }

    if (row0 + 16 <= N) {
#pragma unroll
        for (int t = 0; t < 6; ++t)
#pragma unroll
            for (int r = 0; r < 8; ++r)
                xh2[(size_t)(row0 + r + half * 8) * F2 + t * 16 + l16] = acc[t][r];
    } else {
#pragma unroll
        for (int t = 0; t < 6; ++t)
#pragma unroll
            for (int r = 0; r < 8; ++r) {
                const int m = row0 + r + half * 8;
                if (m < N) xh2[(size_t)m * F2 + t * 16 + l16] = acc[t][r];
            }
    }
}

// ---------------------------------------------------------------------------
// Attention scores
// ---------------------------------------------------------------------------
__global__ void attn_scores1(const float* __restrict__ xh,
                             const float* __restrict__ att_src,
                             const float* __restrict__ att_dst,
                             float* __restrict__ a_src, float* __restrict__ a_dst, int N) {
    const int i = blockIdx.x * blockDim.x + threadIdx.x;   // node*8 + head
    if (i >= N * H1) return;
    const int n = i >> 3, hh = i & 7;
    const float* v  = xh + (size_t)n * F1 + hh * C1;
    const float* as = att_src + hh * C1;
    const float* ad = att_dst + hh * C1;
    float s0 = 0.f, s1 = 0.f;
#pragma unroll 4
    for (int cidx = 0; cidx < C1; ++cidx) {
        const float t = v[cidx];
        s0 += t * as[cidx];
        s1 += t * ad[cidx];
    }
    a_src[i] = s0;
    a_dst[i] = s1;
}

__global__ void attn_scores2(const float* __restrict__ xh2,
                             const float* __restrict__ att_src,
                             const float* __restrict__ att_dst,
                             float* __restrict__ a_src, float* __restrict__ a_dst, int N) {
    const int n = blockIdx.x * blockDim.x + threadIdx.x;
    if (n >= N) return;
    const float* v = xh2 + (size_t)n * F2;
    float s0 = 0.f, s1 = 0.f;
#pragma unroll 4
    for (int cidx = 0; cidx < F2; ++cidx) {
        const float t = v[cidx];
        s0 += t * att_src[cidx];
        s1 += t * att_dst[cidx];
    }
    a_src[n] = s0;
    a_dst[n] = s1;
}

// ---------------------------------------------------------------------------
__global__ void fill_kernel(float* __restrict__ p, long n, float v) {
    const long i = (long)blockIdx.x * blockDim.x + threadIdx.x;
    if (i < n) p[i] = v;
}

__device__ __forceinline__ void edge_sd(const long long* ei, long idx, int E, int& s, int& d) {
    if (idx < (long)E) { s = (int)ei[idx]; d = (int)ei[(long)E + idx]; }
    else               { s = d = (int)(idx - E); }
}

__device__ __forceinline__ void atomic_max_f32(float* p, float v) {
    // ordered-int trick: valid with init = -inf
    if (v >= 0.f) atomicMax((int*)p, __float_as_int(v));
    else          atomicMin((unsigned int*)p, (unsigned int)__float_as_int(v));
}

__device__ __forceinline__ void load8(const float* __restrict__ p, float* o) {
    const float4 u = *(const float4*)p;          // global_load_b128
    const float4 v = *(const float4*)(p + 4);
    o[0] = u.x; o[1] = u.y; o[2] = u.z; o[3] = u.w;
    o[4] = v.x; o[5] = v.y; o[6] = v.z; o[7] = v.w;
}

// ---------------------------------------------------------------------------
// Layer-1 edge passes (H=8)
// ---------------------------------------------------------------------------
__global__ void edge_max1(const long long* __restrict__ ei, int E, int N,
                          const float* __restrict__ a_src, const float* __restrict__ a_dst,
                          float* __restrict__ emax) {
    const long idx = (long)blockIdx.x * blockDim.x + threadIdx.x;
    if (idx >= (long)E + N) return;
    int s, d; edge_sd(ei, idx, E, s, d);
    float es[H1], ed[H1];
    load8(a_src + (size_t)s * H1, es);
    load8(a_dst + (size_t)d * H1, ed);
#pragma unroll
    for (int hh = 0; hh < H1; ++hh) {
        float e = es[hh] + ed[hh];
        e = (e > 0.f) ? e : NEG_SLOPE * e;
        atomic_max_f32(&emax[(size_t)d * H1 + hh], e);
    }
}

__global__ void edge_sum1(const long long* __restrict__ ei, int E, int N,
                          const float* __restrict__ a_src, const float* __restrict__ a_dst,
                          const float* __restrict__ emax, float* __restrict__ denom) {
    const long idx = (long)blockIdx.x * blockDim.x + threadIdx.x;
    if (idx >= (long)E + N) return;
    int s, d; edge_sd(ei, idx, E, s, d);
    float es[H1], ed[H1], em[H1];
    load8(a_src + (size_t)s * H1, es);
    load8(a_dst + (size_t)d * H1, ed);
    load8(emax  + (size_t)d * H1, em);
#pragma unroll
    for (int hh = 0; hh < H1; ++hh) {
        float e = es[hh] + ed[hh];
        e = (e > 0.f) ? e : NEG_SLOPE * e;
        atomicAdd(&denom[(size_t)d * H1 + hh], __expf(e - em[hh]));
    }
}

// 128-thread block per edge; coef[8] via LDS; b128 feature loads (60 % 4 == 0
// -> each aligned 4-float chunk lies within a single head)
__global__ __launch_bounds__(128) void edge_agg1(const long long* __restrict__ ei, int E, int N,
                                                 const float* __restrict__ a_src,
                                                 const float* __restrict__ a_dst,
                                                 const float* __restrict__ emax,
                                                 const float* __restrict__ denom,
                                                 const float* __restrict__ xh,
                                                 float* __restrict__ agg) {
    int s, d; edge_sd(ei, (long)blockIdx.x, E, s, d);
    __shared__ float coef[H1];
    if (threadIdx.x < H1) {
        const int hh = threadIdx.x;
        float e = a_src[(size_t)s * H1 + hh] + a_dst[(size_t)d * H1 + hh];
        e = (e > 0.f) ? e : NEG_SLOPE * e;
        coef[hh] = __expf(e - emax[(size_t)d * H1 + hh]) /
                   (denom[(size_t)d * H1 + hh] + 1e-16f);
    }
    __syncthreads();
    const int c4 = threadIdx.x * 4;          // 0..508; 480 used
    if (c4 < F1) {
        const float4 v = *(const float4*)(xh + (size_t)s * F1 + c4);
        const float  k = coef[c4 / C1];
        float* o = agg + (size_t)d * F1 + c4;
        atomicAdd(o + 0, k * v.x);
        atomicAdd(o + 1, k * v.y);
        atomicAdd(o + 2, k * v.z);
        atomicAdd(o + 3, k * v.w);
    }
}

// h = elu(agg + b1) ; grid = N, block = 480
__global__ __launch_bounds__(F1) void finalize1(const float* __restrict__ agg,
                                                const float* __restrict__ b1,
                                                float* __restrict__ h) {
    const size_t i = (size_t)blockIdx.x * F1 + threadIdx.x;
    const float v = agg[i] + b1[threadIdx.x];
    h[i] = (v > 0.f) ? v : (__expf(v) - 1.f);   // ELU(alpha=1)
}

// ---------------------------------------------------------------------------
// Layer-2 edge passes (H=1, C=96)
// ---------------------------------------------------------------------------
__global__ void edge_max2(const long long* __restrict__ ei, int E, int N,
                          const float* __restrict__ a_src, const float* __restrict__ a_dst,
                          float* __restrict__ emax) {
    const long idx = (long)blockIdx.x * blockDim.x + threadIdx.x;
    if (idx >= (long)E + N) return;
    int s, d; edge_sd(ei, idx, E, s, d);
    float e = a_src[s] + a_dst[d];
    e = (e > 0.f) ? e : NEG_SLOPE * e;
    atomic_max_f32(&emax[d], e);
}

__global__ void edge_sum2(const long long* __restrict__ ei, int E, int N,
                          const float* __restrict__ a_src, const float* __restrict__ a_dst,
                          const float* __restrict__ emax, float* __restrict__ denom) {
    const long idx = (long)blockIdx.x * blockDim.x + threadIdx.x;
    if (idx >= (long)E + N) return;
    int s, d; edge_sd(ei, idx, E, s, d);
    float e = a_src[s] + a_dst[d];
    e = (e > 0.f) ? e : NEG_SLOPE * e;
    atomicAdd(&denom[d], __expf(e - emax[d]));
}

// one wave per edge; threads 0..23 move float4 chunks of the 96-wide row
__global__ __launch_bounds__(32) void edge_agg2(const long long* __restrict__ ei, int E, int N,
                                                const float* __restrict__ a_src,
                                                const float* __restrict__ a_dst,
                                                const float* __restrict__ emax,
                                                const float* __restrict__ denom,
                                                const float* __restrict__ xh2,
                                                float* __restrict__ agg) {
    int s, d; edge_sd(ei, (long)blockIdx.x, E, s, d);
    float e = a_src[s] + a_dst[d];
    e = (e > 0.f) ? e : NEG_SLOPE * e;
    const float coef = __expf(e - emax[d]) / (denom[d] + 1e-16f);
    const int c4 = threadIdx.x * 4;          // 0..124; 96 used
    if (c4 < F2) {
        const float4 v = *(const float4*)(xh2 + (size_t)s * F2 + c4);
        float* o = agg + (size_t)d * F2 + c4;
        atomicAdd(o + 0, coef * v.x);
        atomicAdd(o + 1, coef * v.y);
        atomicAdd(o + 2, coef * v.z);
        atomicAdd(o + 3, coef * v.w);
    }
}

// out = agg2 + b2 ; grid = N, block = 96
__global__ __launch_bounds__(F2) void finalize2(const float* __restrict__ agg,
                                                const float* __restrict__ b2,
                                                float* __restrict__ out) {
    const size_t i = (size_t)blockIdx.x * F2 + threadIdx.x;
    out[i] = agg[i] + b2[threadIdx.x];
}

// ---------------------------------------------------------------------------
extern "C" void kernel_launch(void* const* d_in, const int* in_sizes, int n_in,
                              void* d_out, int out_size, void* d_ws, size_t ws_size,
                              hipStream_t stream) {
    const float*     x        = (const float*)d_in[0];
    const long long* ei       = (const long long*)d_in[1];   // int64 [2,E]
    const float*     W1       = (const float*)d_in[2];
    const float*     att_src1 = (const float*)d_in[3];
    const float*     att_dst1 = (const float*)d_in[4];
    const float*     b1       = (const float*)d_in[5];
    const float*     W2       = (const float*)d_in[6];
    const float*     att_src2 = (const float*)d_in[7];
    const float*     att_dst2 = (const float*)d_in[8];
    const float*     b2       = (const float*)d_in[9];
    float*           out      = (float*)d_out;

    const int  N    = in_sizes[0] / 4;     // x is [N,4]
    const int  E    = in_sizes[1] / 2;     // edge_index is [2,E]
    const long Etot = (long)E + N;         // + self loops

    // workspace layout (floats)
    float* bufA   = (float*)d_ws;                 // N*480: xh1, then h (overwritten)
    float* bufB   = bufA + (size_t)N * F1;        // N*480: agg1; later xh2 | agg2
    float* sm     = bufB + (size_t)N * F1;
    float* a_src1 = sm;                            // N*8
    float* a_dst1 = a_src1 + (size_t)N * H1;       // N*8
    float* emax1  = a_dst1 + (size_t)N * H1;       // N*8
    float* denom1 = emax1  + (size_t)N * H1;       // N*8
    float* a_src2 = denom1 + (size_t)N * H1;       // N
    float* a_dst2 = a_src2 + N;                    // N
    float* emax2  = a_dst2 + N;                    // N
    float* denom2 = emax2  + N;                    // N

    float* xh2  = bufB;                     // N*96  (agg1 dead after finalize1)
    float* agg2 = bufB + (size_t)N * F2;    // N*96

    const int TB = 256;
    const long eg = (Etot + TB - 1) / TB;

    // ---- layer 1 ----
    gemm1_wmma<<<dim3((N + 15) / 16, F1 / 16), 32, 0, stream>>>(x, W1, bufA, N);
    attn_scores1<<<((long)N * H1 + TB - 1) / TB, TB, 0, stream>>>(bufA, att_src1, att_dst1,
                                                                  a_src1, a_dst1, N);
    fill_kernel<<<((long)N * H1 + TB - 1) / TB, TB, 0, stream>>>(emax1, (long)N * H1, -INFINITY);
    fill_kernel<<<((long)N * H1 + TB - 1) / TB, TB, 0, stream>>>(denom1, (long)N * H1, 0.f);
    fill_kernel<<<((long)N * F1 + TB - 1) / TB, TB, 0, stream>>>(bufB, (long)N * F1, 0.f);
    edge_max1<<<eg, TB, 0, stream>>>(ei, E, N, a_src1, a_dst1, emax1);
    edge_sum1<<<eg, TB, 0, stream>>>(ei, E, N, a_src1, a_dst1, emax1, denom1);
    edge_agg1<<<(unsigned)Etot, 128, 0, stream>>>(ei, E, N, a_src1, a_dst1, emax1, denom1,
                                                  bufA, bufB);
    finalize1<<<N, F1, 0, stream>>>(bufB, b1, bufA);

    // ---- layer 2 ----
    gemm2_wmma<<<(N + 15) / 16, 32, 0, stream>>>(bufA, W2, xh2, N);
    attn_scores2<<<(N + TB - 1) / TB, TB, 0, stream>>>(xh2, att_src2, att_dst2,
                                                       a_src2, a_dst2, N);
    fill_kernel<<<(N + TB - 1) / TB, TB, 0, stream>>>(emax2, N, -INFINITY);
    fill_kernel<<<(N + TB - 1) / TB, TB, 0, stream>>>(denom2, N, 0.f);
    fill_kernel<<<((long)N * F2 + TB - 1) / TB, TB, 0, stream>>>(agg2, (long)N * F2, 0.f);
    edge_max2<<<eg, TB, 0, stream>>>(ei, E, N, a_src2, a_dst2, emax2);
    edge_sum2<<<eg, TB, 0, stream>>>(ei, E, N, a_src2, a_dst2, emax2, denom2);
    edge_agg2<<<(unsigned)Etot, 32, 0, stream>>>(ei, E, N, a_src2, a_dst2, emax2, denom2,
                                                 xh2, agg2);
    finalize2<<<N, F2, 0, stream>>>(agg2, b2, out);
}